// ComprehensiveLoss_16329465659791
// MI455X (gfx1250) — compile-verified
//
#include <hip/hip_runtime.h>
#include <math.h>

#define TPB   256
#define NBLK  1024
#define IMG_W 512
#define IMG_H 512
#define NTOT  (32 * 512 * 512)
#define NQ    12

typedef __attribute__((ext_vector_type(2))) float v2f;
typedef __attribute__((ext_vector_type(8))) float v8f;

// ---- wave32 sum via V_WMMA_F32_16X16X4_F32 (B = ones) ----------------------
// A (16x4 f32) holds each lane's value in both of its 2 VGPR slots, so
// sum_{m,k} A[m,k] = 2 * sum_lanes(v).  With B = ones, D[m][n] = rowsum(m)
// for every n; lanes 0-15 hold one half of the rows in their 8 C VGPRs and
// lanes 16-31 the other half, so (lane sum + cross-half shuffle) * 0.5 is the
// exact f32 wave total.  EXEC must be all ones at call sites (it is: all
// callers are uniform full-block code paths).
__device__ __forceinline__ float wave_sum32(float v) {
#if defined(__gfx1250__) && __has_builtin(__builtin_amdgcn_wmma_f32_16x16x4_f32)
  v2f a; a.x = v;    a.y = v;
  v2f b; b.x = 1.0f; b.y = 1.0f;
  v8f c = {};
  c = __builtin_amdgcn_wmma_f32_16x16x4_f32(false, a, false, b, (short)0, c,
                                            false, false);
  float s = c[0] + c[1] + c[2] + c[3] + c[4] + c[5] + c[6] + c[7];
  s += __shfl_down(s, 16, 32);
  return s * 0.5f;
#else
  for (int o = 16; o > 0; o >>= 1) v += __shfl_down(v, o, 32);
  return v;
#endif
}

// Deterministic block sum (valid on thread 0). 256 threads = 8 wave32s.
__device__ __forceinline__ float block_sum(float v, float* lds) {
  float w = wave_sum32(v);
  int lane = threadIdx.x & 31;
  int wid  = threadIdx.x >> 5;
  if (lane == 0) lds[wid] = w;
  __syncthreads();
  float r = 0.0f;
  if (threadIdx.x == 0) {
    for (int k = 0; k < (int)(blockDim.x >> 5); ++k) r += lds[k];
  }
  __syncthreads();
  return r;
}

// mode: 0 = raw float buffer, 1 = sigmoid(float logits), 2 = int -> float
__device__ __forceinline__ float decode(const void* src, int mode, int i) {
  if (mode == 2) return (float)((const int*)src)[i];
  float x = ((const float*)src)[i];
  if (mode == 1) return 1.0f / (1.0f + expf(-x));
  return x;
}

// ---- fused pointwise + boundary-weighted BCE reductions (rows 0..5) --------
// 0: sum bce  1: sum p  2: sum t  3: sum p*t  4: sum (p>.5)  5: sum w*bce
__global__ void base_reduce_kernel(const float* __restrict__ pred,
                                   const int* __restrict__ target,
                                   float* __restrict__ partials, int total) {
  __shared__ float lds[TPB / 32];
  float a0 = 0, a1 = 0, a2 = 0, a3 = 0, a4 = 0, a5 = 0;
  for (int i = blockIdx.x * blockDim.x + threadIdx.x; i < total;
       i += gridDim.x * blockDim.x) {
    float x = pred[i];
    float t = (float)target[i];
    float p = 1.0f / (1.0f + expf(-x));
    float bce = fmaxf(x, 0.0f) - x * t + log1pf(expf(-fabsf(x)));
    int w = i & (IMG_W - 1);
    int h = (i >> 9) & (IMG_H - 1);
    int h0 = (h > 0) ? -1 : 0, h1 = (h < IMG_H - 1) ? 1 : 0;
    int w0 = (w > 0) ? -1 : 0, w1 = (w < IMG_W - 1) ? 1 : 0;
    int tmin = target[i], tmax = tmin;
    for (int dh = h0; dh <= h1; ++dh)
      for (int dw = w0; dw <= w1; ++dw) {
        int tv = target[i + dh * IMG_W + dw];
        tmin = min(tmin, tv);
        tmax = max(tmax, tv);
      }
    float wgt = 1.0f + 3.0f * (float)(tmax - tmin);  // 1 + 3*(dilate - erode)
    a0 += bce;
    a1 += p;
    a2 += t;
    a3 += p * t;
    a4 += (p > 0.5f) ? 1.0f : 0.0f;
    a5 += wgt * bce;
  }
  float r;
  r = block_sum(a0, lds); if (threadIdx.x == 0) partials[0 * NBLK + blockIdx.x] = r;
  r = block_sum(a1, lds); if (threadIdx.x == 0) partials[1 * NBLK + blockIdx.x] = r;
  r = block_sum(a2, lds); if (threadIdx.x == 0) partials[2 * NBLK + blockIdx.x] = r;
  r = block_sum(a3, lds); if (threadIdx.x == 0) partials[3 * NBLK + blockIdx.x] = r;
  r = block_sum(a4, lds); if (threadIdx.x == 0) partials[4 * NBLK + blockIdx.x] = r;
  r = block_sum(a5, lds); if (threadIdx.x == 0) partials[5 * NBLK + blockIdx.x] = r;
}

// ---- soft_erode = min over cross (valid-window clamped) --------------------
__global__ void cross_min_kernel(const void* __restrict__ src, int mode,
                                 float* __restrict__ dst, int total) {
  int i = blockIdx.x * blockDim.x + threadIdx.x;
  if (i >= total) return;
  int w = i & (IMG_W - 1);
  int h = (i >> 9) & (IMG_H - 1);
  float m = decode(src, mode, i);
  if (w > 0)         m = fminf(m, decode(src, mode, i - 1));
  if (w < IMG_W - 1) m = fminf(m, decode(src, mode, i + 1));
  if (h > 0)         m = fminf(m, decode(src, mode, i - IMG_W));
  if (h < IMG_H - 1) m = fminf(m, decode(src, mode, i + IMG_W));
  dst[i] = m;
}

// ---- fused dilate3(e) + skeleton init/update (open never materialized) -----
// init:  skel = relu(ref - dilate3(e))
// update: delta = relu(ref - dilate3(e)); skel += relu(delta - skel*delta)
__global__ void open_update_kernel(const void* __restrict__ ref, int ref_mode,
                                   const float* __restrict__ e,
                                   float* __restrict__ skel, int init_flag,
                                   int total) {
  int i = blockIdx.x * blockDim.x + threadIdx.x;
  if (i >= total) return;
  int w = i & (IMG_W - 1);
  int h = (i >> 9) & (IMG_H - 1);
  int h0 = (h > 0) ? -1 : 0, h1 = (h < IMG_H - 1) ? 1 : 0;
  int w0 = (w > 0) ? -1 : 0, w1 = (w < IMG_W - 1) ? 1 : 0;
  float o = -INFINITY;
  for (int dh = h0; dh <= h1; ++dh)
    for (int dw = w0; dw <= w1; ++dw)
      o = fmaxf(o, e[i + dh * IMG_W + dw]);
  float r = decode(ref, ref_mode, i);
  float d = fmaxf(r - o, 0.0f);
  if (init_flag) {
    skel[i] = d;
  } else {
    float s = skel[i];
    skel[i] = s + fmaxf(d - s * d, 0.0f);
  }
}

// rows: sum(skel * decode(other)), sum(skel)
__global__ void skel_reduce_kernel(const float* __restrict__ skel,
                                   const void* __restrict__ other, int mode,
                                   float* __restrict__ partials, int rowA,
                                   int rowB, int total) {
  __shared__ float lds[TPB / 32];
  float aa = 0.0f, ab = 0.0f;
  for (int i = blockIdx.x * blockDim.x + threadIdx.x; i < total;
       i += gridDim.x * blockDim.x) {
    float s = skel[i];
    aa += s * decode(other, mode, i);
    ab += s;
  }
  float r;
  r = block_sum(aa, lds); if (threadIdx.x == 0) partials[rowA * NBLK + blockIdx.x] = r;
  r = block_sum(ab, lds); if (threadIdx.x == 0) partials[rowB * NBLK + blockIdx.x] = r;
}

// ---- distance transform ----------------------------------------------------
__global__ void dt_init_kernel(const float* __restrict__ pred,
                               const int* __restrict__ target, int from_pred,
                               float* __restrict__ cur,
                               float* __restrict__ dist, int total) {
  int i = blockIdx.x * blockDim.x + threadIdx.x;
  if (i >= total) return;
  float m;
  if (from_pred) {
    float p = 1.0f / (1.0f + expf(-pred[i]));
    m = (p > 0.5f) ? 0.0f : 1.0f;  // 1 - pred_binary
  } else {
    m = target[i] ? 0.0f : 1.0f;   // 1 - t
  }
  cur[i] = m;
  dist[i] = 0.0f;
}

__global__ void dt_step_kernel(const float* __restrict__ cur,
                               float* __restrict__ nxt,
                               float* __restrict__ dist, float dval,
                               int total) {
  int i = blockIdx.x * blockDim.x + threadIdx.x;
  if (i >= total) return;
  int w = i & (IMG_W - 1);
  int h = (i >> 9) & (IMG_H - 1);
  int h0 = (h > 0) ? -1 : 0, h1 = (h < IMG_H - 1) ? 1 : 0;
  int w0 = (w > 0) ? -1 : 0, w1 = (w < IMG_W - 1) ? 1 : 0;
  float e = INFINITY;
  for (int dh = h0; dh <= h1; ++dh)
    for (int dw = w0; dw <= w1; ++dw)
      e = fminf(e, cur[i + dh * IMG_W + dw]);
  nxt[i] = e;
  float c = cur[i];
  if (c > 0.5f && e < 0.5f && dist[i] == 0.0f) dist[i] = dval;
}

// final-dist fixup folded into reduction: sum(dist_final * weight)
__global__ void dt_reduce_kernel(const float* __restrict__ dist,
                                 const float* __restrict__ pred,
                                 const int* __restrict__ target,
                                 int is_pred_dt, float* __restrict__ partials,
                                 int row, int total) {
  __shared__ float lds[TPB / 32];
  float acc = 0.0f;
  for (int i = blockIdx.x * blockDim.x + threadIdx.x; i < total;
       i += gridDim.x * blockDim.x) {
    float p  = 1.0f / (1.0f + expf(-pred[i]));
    float pb = (p > 0.5f) ? 1.0f : 0.0f;
    float t  = (float)target[i];
    float m, wgt;
    if (is_pred_dt) { m = 1.0f - pb; wgt = t; }
    else            { m = 1.0f - t;  wgt = pb; }
    float dv = dist[i];
    float df = ((m > 0.5f) && (dv == 0.0f)) ? 20.0f : dv;
    acc += df * wgt;
  }
  float r = block_sum(acc, lds);
  if (threadIdx.x == 0) partials[row * NBLK + blockIdx.x] = r;
}

// ---- final combine ---------------------------------------------------------
__global__ void final_kernel(const float* __restrict__ partials,
                             float* __restrict__ out) {
  __shared__ float lds[TPB / 32];
  __shared__ float sums[NQ];
  for (int q = 0; q < NQ; ++q) {
    float v = 0.0f;
    for (int j = threadIdx.x; j < NBLK; j += blockDim.x)
      v += partials[q * NBLK + j];
    float s = block_sum(v, lds);
    if (threadIdx.x == 0) sums[q] = s;
    __syncthreads();
  }
  if (threadIdx.x == 0) {
    const float Nf = (float)NTOT;
    float sum_bce = sums[0], sum_p = sums[1], sum_t = sums[2];
    float inter = sums[3], sum_pb = sums[4], sum_wbce = sums[5];
    float sp_t = sums[6], sp = sums[7], st_p = sums[8], st = sums[9];
    float hdf = sums[10], hdb = sums[11];

    float loss_bce = sum_bce / Nf;
    float loss_dice = 1.0f - (2.0f * inter + 1.0f) / (sum_p + sum_t + 1.0f);
    float fp = sum_p - inter, fn = sum_t - inter;
    float tversky = (inter + 1.0f) / (inter + 0.3f * fp + 0.7f * fn + 1.0f);
    float loss_ft = powf(1.0f - tversky, 1.33f);
    float loss_boundary = sum_wbce / Nf;
    float tprec = (sp_t + 1.0f) / (sp + 1.0f);
    float tsens = (st_p + 1.0f) / (st + 1.0f);
    float loss_cldice = 1.0f - 2.0f * tprec * tsens / (tprec + tsens);
    float hd_fwd = (hdf + 1e-6f) / (sum_t + 1e-6f);
    float hd_bwd = (hdb + 1e-6f) / (sum_pb + 1e-6f);
    float loss_hd = 0.5f * (hd_fwd + hd_bwd);

    out[0] = 0.2f * loss_bce + 0.2f * loss_dice + 0.2f * loss_cldice +
             0.1f * loss_hd + 0.1f * loss_boundary + 0.2f * loss_ft;
  }
}

// ---- host driver -----------------------------------------------------------
extern "C" void kernel_launch(void* const* d_in, const int* in_sizes, int n_in,
                              void* d_out, int out_size, void* d_ws,
                              size_t ws_size, hipStream_t stream) {
  (void)in_sizes; (void)n_in; (void)out_size; (void)ws_size;
  const float* pred = (const float*)d_in[0];
  const int* target = (const int*)d_in[1];
  float* out = (float*)d_out;
  const int total = NTOT;

  float* b1 = (float*)d_ws;        // ping / DT cur
  float* b2 = b1 + total;          // pong / DT dist
  float* b3 = b2 + total;          // skel  / DT nxt
  float* partials = b3 + total;    // NQ x NBLK floats

  dim3 sgrid(total / TPB), sblk(TPB);
  dim3 rgrid(NBLK), rblk(TPB);

  // rows 0..5: BCE/Dice/Tversky sums + boundary-weighted BCE (fused stencil)
  base_reduce_kernel<<<rgrid, rblk, 0, stream>>>(pred, target, partials, total);

  // ---- skeleton of pred_prob (mode 1) ----
  cross_min_kernel<<<sgrid, sblk, 0, stream>>>((const void*)pred, 1, b1, total);
  open_update_kernel<<<sgrid, sblk, 0, stream>>>((const void*)pred, 1, b1, b3,
                                                 1, total);
  {
    float* cur = b1; float* nxt = b2;
    for (int it = 0; it < 10; ++it) {
      cross_min_kernel<<<sgrid, sblk, 0, stream>>>((const void*)cur, 0, nxt,
                                                   total);
      open_update_kernel<<<sgrid, sblk, 0, stream>>>((const void*)cur, 0, nxt,
                                                     b3, 0, total);
      float* tmp = cur; cur = nxt; nxt = tmp;
    }
  }
  skel_reduce_kernel<<<rgrid, rblk, 0, stream>>>(b3, (const void*)target, 2,
                                                 partials, 6, 7, total);

  // ---- skeleton of t (mode 2) ----
  cross_min_kernel<<<sgrid, sblk, 0, stream>>>((const void*)target, 2, b1,
                                               total);
  open_update_kernel<<<sgrid, sblk, 0, stream>>>((const void*)target, 2, b1,
                                                 b3, 1, total);
  {
    float* cur = b1; float* nxt = b2;
    for (int it = 0; it < 10; ++it) {
      cross_min_kernel<<<sgrid, sblk, 0, stream>>>((const void*)cur, 0, nxt,
                                                   total);
      open_update_kernel<<<sgrid, sblk, 0, stream>>>((const void*)cur, 0, nxt,
                                                     b3, 0, total);
      float* tmp = cur; cur = nxt; nxt = tmp;
    }
  }
  skel_reduce_kernel<<<rgrid, rblk, 0, stream>>>(b3, (const void*)pred, 1,
                                                 partials, 8, 9, total);

  // ---- distance transform of (1 - pred_binary), 20 fused erode+update ----
  dt_init_kernel<<<sgrid, sblk, 0, stream>>>(pred, target, 1, b1, b2, total);
  {
    float* cur = b1; float* nxt = b3;
    for (int d = 1; d <= 20; ++d) {
      dt_step_kernel<<<sgrid, sblk, 0, stream>>>(cur, nxt, b2, (float)d,
                                                 total);
      float* tmp = cur; cur = nxt; nxt = tmp;
    }
  }
  dt_reduce_kernel<<<rgrid, rblk, 0, stream>>>(b2, pred, target, 1, partials,
                                               10, total);

  // ---- distance transform of (1 - t) ----
  dt_init_kernel<<<sgrid, sblk, 0, stream>>>(pred, target, 0, b1, b2, total);
  {
    float* cur = b1; float* nxt = b3;
    for (int d = 1; d <= 20; ++d) {
      dt_step_kernel<<<sgrid, sblk, 0, stream>>>(cur, nxt, b2, (float)d,
                                                 total);
      float* tmp = cur; cur = nxt; nxt = tmp;
    }
  }
  dt_reduce_kernel<<<rgrid, rblk, 0, stream>>>(b2, pred, target, 0, partials,
                                               11, total);

  // ---- combine scalars deterministically ----
  final_kernel<<<1, TPB, 0, stream>>>(partials, out);
}